// RegionProposalNetwork_71322226917405
// MI455X (gfx1250) — compile-verified
//
#include <hip/hip_runtime.h>
#include <stdint.h>

// ---------------- problem constants (img 2048x2048, strides 4..64, 3 anchors) ----
#define KSEL   1000                 // PRE/POST_NMS_TOP_N
#define NLVL   5
#define MTOT   (KSEL * NLVL)        // 5000 candidates after per-level top-k
#define NW     ((MTOT + 63) / 64)   // 79 mask words per row
#define EQCAP  2048
#define ATOT   1047552              // total anchors
#define IMGW   2048.0f
#define IMGH   2048.0f
#define BBOX_CLIP_F 4.135166556742356f   // log(1000/16)
#define NMS_TH 0.7f

// state[] layout: [0..4]=prefix, [5..9]=remaining(E), [10..14]=selCount(G),
//                 [15..19]=eqCount, [20]=max-coordinate bits
#define ST_PFX 0
#define ST_REM 5
#define ST_SEL 10
#define ST_EQ  15
#define ST_MAX 20

__device__ __host__ __forceinline__ int lvl_off(int l) {
  switch (l) {
    case 0: return 0;
    case 1: return 786432;
    case 2: return 983040;
    case 3: return 1032192;
    case 4: return 1044480;
    default: return 1047552;
  }
}

__device__ __forceinline__ int level_of(int ga) {
  if (ga < 786432)  return 0;
  if (ga < 983040)  return 1;
  if (ga < 1032192) return 2;
  if (ga < 1044480) return 3;
  return 4;
}

// monotonic float -> uint key (larger float => larger key)
__device__ __forceinline__ unsigned int fkey(float f) {
  unsigned int u = __float_as_uint(f);
  return (u & 0x80000000u) ? ~u : (u | 0x80000000u);
}

// decode + clip one box, exactly mirroring the reference arithmetic order
__device__ __forceinline__ float4 decode_clip(const float* __restrict__ deltas,
                                              const float* __restrict__ anchors,
                                              int ga, int* valid) {
  float ax1 = anchors[ga * 4 + 0], ay1 = anchors[ga * 4 + 1];
  float ax2 = anchors[ga * 4 + 2], ay2 = anchors[ga * 4 + 3];
  float w = ax2 - ax1, h = ay2 - ay1;
  float cx = ax1 + 0.5f * w, cy = ay1 + 0.5f * h;
  float dx = deltas[ga * 4 + 0], dy = deltas[ga * 4 + 1];
  float dw = fminf(deltas[ga * 4 + 2], BBOX_CLIP_F);
  float dh = fminf(deltas[ga * 4 + 3], BBOX_CLIP_F);
  float pcx = dx * w + cx, pcy = dy * h + cy;
  float pw = expf(dw) * w, ph = expf(dh) * h;
  float x1 = pcx - 0.5f * pw, y1 = pcy - 0.5f * ph;
  float x2 = pcx + 0.5f * pw, y2 = pcy + 0.5f * ph;
  x1 = fminf(fmaxf(x1, 0.0f), IMGW);
  y1 = fminf(fmaxf(y1, 0.0f), IMGH);
  x2 = fminf(fmaxf(x2, 0.0f), IMGW);
  y2 = fminf(fmaxf(y2, 0.0f), IMGH);
  *valid = (x2 - x1 >= 0.001f) && (y2 - y1 >= 0.001f);
  return make_float4(x1, y1, x2, y2);
}

// ---------------- kernel 0: re-init everything we read (ws is poisoned once) ----
__global__ void k_init(unsigned int* hist, unsigned int* state,
                       unsigned long long* validBits, float* out) {
  int stride = gridDim.x * blockDim.x;
  int i0 = blockIdx.x * blockDim.x + threadIdx.x;
  for (int t = i0; t < 4 * NLVL * 256; t += stride) hist[t] = 0u;
  for (int t = i0; t < 4 * KSEL; t += stride) out[t] = 0.0f;
  for (int t = i0; t < NW; t += stride) validBits[t] = 0ull;
  if (i0 < NLVL) {
    state[ST_PFX + i0] = 0u;
    int n = lvl_off(i0 + 1) - lvl_off(i0);
    state[ST_REM + i0] = (unsigned int)(n < KSEL ? n : KSEL);
    state[ST_SEL + i0] = 0u;
    state[ST_EQ + i0]  = 0u;
  }
  if (i0 == NLVL) state[ST_MAX] = 0u;  // +0.0f
}

// ---------------- radix-select pass: per-level 256-bin histogram ----------------
__global__ void k_hist(const float* __restrict__ obj, const unsigned int* __restrict__ state,
                       unsigned int* histp, int shiftB, unsigned int pmask) {
  __shared__ unsigned int lh[256];
  int lvl = blockIdx.y;
  for (int t = threadIdx.x; t < 256; t += blockDim.x) lh[t] = 0u;
  __syncthreads();
  int off = lvl_off(lvl);
  int n = lvl_off(lvl + 1) - off;
  unsigned int pfx = state[ST_PFX + lvl] & pmask;
  int stride = gridDim.x * blockDim.x;
  for (int t = blockIdx.x * blockDim.x + threadIdx.x; t < n; t += stride) {
    unsigned int key = fkey(obj[off + t]);
    if ((key & pmask) == pfx) atomicAdd(&lh[(key >> shiftB) & 0xFFu], 1u);
  }
  __syncthreads();
  for (int t = threadIdx.x; t < 256; t += blockDim.x) {
    unsigned int v = lh[t];
    if (v) atomicAdd(&histp[lvl * 256 + t], v);
  }
}

// pick the bin containing the k-th largest; refine prefix & remaining
__global__ void k_select(const unsigned int* __restrict__ histp, unsigned int* state, int shiftB) {
  if (threadIdx.x != 0) return;
  int lvl = blockIdx.x;
  const unsigned int* h = histp + lvl * 256;
  unsigned int rem = state[ST_REM + lvl];
  unsigned int cum = 0;
  int b = 255;
  for (; b > 0; --b) {
    unsigned int cn = h[b];
    if (cum + cn >= rem) break;
    cum += cn;
  }
  state[ST_PFX + lvl] |= ((unsigned int)b) << shiftB;
  state[ST_REM + lvl] = rem - cum;   // after last pass: E = #equals needed
}

// ---------------- compaction: gather key>T, stash key==T candidates ------------
__global__ void k_compact(const float* __restrict__ obj, unsigned int* state,
                          unsigned int* sel, unsigned int* eqbuf) {
  int stride = gridDim.x * blockDim.x;
  for (int i = blockIdx.x * blockDim.x + threadIdx.x; i < ATOT; i += stride) {
    int lvl = level_of(i);
    unsigned int T = state[ST_PFX + lvl];
    unsigned int key = fkey(obj[i]);
    if (key > T) {
      unsigned int s = atomicAdd(&state[ST_SEL + lvl], 1u);
      if (s < (unsigned)KSEL) sel[lvl * KSEL + s] = (unsigned int)i;
    } else if (key == T) {
      unsigned int e = atomicAdd(&state[ST_EQ + lvl], 1u);
      if (e < (unsigned)EQCAP) eqbuf[lvl * EQCAP + e] = (unsigned int)i;
    }
  }
}

// resolve threshold ties lowest-index-first (matches jax.lax.top_k)
__global__ void k_eq(unsigned int* state, const unsigned int* __restrict__ eqbuf,
                     unsigned int* sel) {
  int lvl = blockIdx.x;
  unsigned int G = state[ST_SEL + lvl];
  unsigned int E = state[ST_REM + lvl];
  unsigned int c = state[ST_EQ + lvl];
  if (c > (unsigned)EQCAP) c = EQCAP;
  for (unsigned int t = threadIdx.x; t < c; t += blockDim.x) {
    unsigned int v = eqbuf[lvl * EQCAP + t];
    unsigned int r = 0;
    for (unsigned int u = 0; u < c; ++u) r += (eqbuf[lvl * EQCAP + u] < v) ? 1u : 0u;
    if (r < E && (G + r) < (unsigned)KSEL) sel[lvl * KSEL + G + r] = v;
  }
}

// ---------------- build 64-bit sort keys; track global max coordinate ----------
__global__ void k_keys(const float* __restrict__ obj, const float* __restrict__ deltas,
                       const float* __restrict__ anchors, const unsigned int* __restrict__ sel,
                       unsigned int* state, unsigned long long* keys) {
  int i = blockIdx.x * blockDim.x + threadIdx.x;
  if (i >= MTOT) return;
  int ga = (int)sel[i];
  int valid;
  float4 b = decode_clip(deltas, anchors, ga, &valid);
  // coords are clipped to [0, 2048] => non-negative => uint-bit max == float max
  atomicMax(&state[ST_MAX], __float_as_uint(b.x));
  atomicMax(&state[ST_MAX], __float_as_uint(b.y));
  atomicMax(&state[ST_MAX], __float_as_uint(b.z));
  atomicMax(&state[ST_MAX], __float_as_uint(b.w));
  unsigned int sk = valid ? fkey(obj[ga]) : 0u;  // invalid => below every valid score
  keys[i] = ((unsigned long long)sk << 20) |
            (unsigned long long)(0xFFFFFu - (unsigned int)ga);  // stable tiebreak: ga asc
}

// ---------------- single-block bitonic sort of 8192 u64 keys in LDS ------------
__global__ void __launch_bounds__(1024) k_sort(const unsigned long long* __restrict__ keysIn,
                                               unsigned int* sortedGa) {
  __shared__ unsigned long long sk[8192];  // 64 KB of the WGP's 320 KB LDS
  int tid = threadIdx.x;
  for (int i = tid; i < 8192; i += 1024) sk[i] = (i < MTOT) ? keysIn[i] : 0ull;
  __syncthreads();
  for (int k = 2; k <= 8192; k <<= 1) {
    for (int j = k >> 1; j > 0; j >>= 1) {
      for (int i = tid; i < 8192; i += 1024) {
        int ixj = i ^ j;
        if (ixj > i) {
          bool asc = (i & k) != 0;  // inverted canonical => overall DESCENDING
          unsigned long long a = sk[i], b = sk[ixj];
          if ((a > b) == asc) { sk[i] = b; sk[ixj] = a; }
        }
      }
      __syncthreads();
    }
  }
  for (int i = tid; i < MTOT; i += 1024)
    sortedGa[i] = 0xFFFFFu - (unsigned int)(sk[i] & 0xFFFFFu);
}

// ---------------- recompute sorted boxes / level-offset boxes / areas -----------
__global__ void k_post(const float* __restrict__ deltas, const float* __restrict__ anchors,
                       const unsigned int* __restrict__ sortedGa,
                       const unsigned int* __restrict__ state,
                       float* boxesS, float4* nbS, float* areasS,
                       unsigned long long* validBits) {
  int i = blockIdx.x * blockDim.x + threadIdx.x;
  if (i >= MTOT) return;
  int ga = (int)sortedGa[i];
  int valid;
  float4 b = decode_clip(deltas, anchors, ga, &valid);
  float maxc = __uint_as_float(state[ST_MAX]);
  float s = (float)level_of(ga) * (maxc + 1.0f);   // per-level NMS offset
  float4 nb = make_float4(b.x + s, b.y + s, b.z + s, b.w + s);
  ((float4*)boxesS)[i] = b;
  nbS[i] = nb;
  areasS[i] = (nb.z - nb.x) * (nb.w - nb.y);       // same arithmetic as reference
  if (valid) atomicOr(&validBits[i >> 6], 1ull << (i & 63));
}

// ---------------- pairwise suppression bitmask (upper triangle) -----------------
__global__ void k_mask(const float4* __restrict__ nbS, const float* __restrict__ areasS,
                       unsigned long long* mask) {
  if (blockIdx.x < blockIdx.y) return;  // lower-triangle words are never consumed
  int cb0 = blockIdx.x * 64;
  int rb0 = blockIdx.y * 64;
  __shared__ float4 cbx[64];
  __shared__ float  cav[64];
  int tid = threadIdx.x;
  int col = cb0 + tid;
#if defined(__gfx1250__)
  // CDNA5 async global->LDS staging of the 64 column boxes (ASYNCcnt path).
  if (col < MTOT) {
    __attribute__((address_space(3))) float4* lds3 =
        (__attribute__((address_space(3))) float4*)cbx;
    unsigned int ldsoff = (unsigned int)(unsigned long long)(lds3 + tid);
    unsigned long long gaddr = (unsigned long long)(nbS + col);
    asm volatile("global_load_async_to_lds_b128 %0, %1, off"
                 :: "v"(ldsoff), "v"(gaddr) : "memory");
    cav[tid] = areasS[col];
  }
#if __has_builtin(__builtin_amdgcn_s_wait_asynccnt)
  __builtin_amdgcn_s_wait_asynccnt(0);
#else
  asm volatile("s_wait_asynccnt 0" ::: "memory");
#endif
#else
  if (col < MTOT) { cbx[tid] = nbS[col]; cav[tid] = areasS[col]; }
#endif
  __syncthreads();

  int r = rb0 + tid;
  if (r >= MTOT) return;
  float4 b = nbS[r];
  float ar = areasS[r];
  unsigned long long bits = 0ull;
  int cmax = (MTOT - cb0) < 64 ? (MTOT - cb0) : 64;
  for (int c = 0; c < cmax; ++c) {
    int j = cb0 + c;
    if (j <= r) continue;                 // only j > r can be suppressed by r
    float4 o = cbx[c];
    float iw = fmaxf(fminf(b.z, o.z) - fmaxf(b.x, o.x), 0.0f);
    float ih = fmaxf(fminf(b.w, o.w) - fmaxf(b.y, o.y), 0.0f);
    float inter = iw * ih;
    float iou = inter / (ar + cav[c] - inter);
    if (iou > NMS_TH) bits |= (1ull << c);
  }
  mask[(size_t)r * NW + blockIdx.x] = bits;
}

// ---------------- greedy NMS: one wave32, bitmask candidate skipping ------------
// Scans words of (validBits & ~removed); __ffsll jumps straight to the next live
// candidate, so the serial loop runs ~(#kept + NW) iterations instead of 5000.
// Only mask words >= current word are fetched/OR'd (earlier columns are already
// resolved). Single-wave workgroup: barriers degrade to S_NOP on CDNA5.
__global__ void k_reduce(const float* __restrict__ boxesS,
                         const unsigned long long* __restrict__ validBits,
                         const unsigned long long* __restrict__ mask, float* out) {
  __shared__ unsigned long long removed[NW];
  int lane = threadIdx.x;  // 32 threads == 1 wave
  for (int w = lane; w < NW; w += 32) removed[w] = 0ull;
  __syncthreads();
  int cnt = 0;
  for (int w = 0; w < NW && cnt < KSEL; ++w) {
    unsigned long long avail = validBits[w] & ~removed[w];
    while (avail) {
      int b = __ffsll((unsigned long long)avail) - 1;
      int i = (w << 6) + b;
      const unsigned long long* row = mask + (size_t)i * NW;
      // prefetch the part of the row we are about to OR (gfx1250 global_prefetch_b8)
      if (lane < 10)
        __builtin_prefetch(((const char*)(row + w)) + lane * 64, 0, 0);
      for (int t = w + lane; t < NW; t += 32) removed[t] |= row[t];
      if (lane < 4) out[cnt * 4 + lane] = boxesS[i * 4 + lane];
      ++cnt;
      if (cnt >= KSEL) break;            // remaining keeps can't change the output
      __syncthreads();
      unsigned long long hi = (b == 63) ? 0ull : (~0ull << (b + 1));
      avail = validBits[w] & ~removed[w] & hi;
    }
    __syncthreads();
  }
}

// ---------------- host driver ---------------------------------------------------
extern "C" void kernel_launch(void* const* d_in, const int* in_sizes, int n_in,
                              void* d_out, int out_size, void* d_ws, size_t ws_size,
                              hipStream_t stream) {
  (void)in_sizes; (void)n_in; (void)out_size; (void)ws_size;
  const float* obj     = (const float*)d_in[0];
  const float* deltas  = (const float*)d_in[1];
  const float* anchors = (const float*)d_in[2];
  float* out = (float*)d_out;

  char* base = (char*)d_ws;
  size_t off = 0;
  auto take = [&](size_t bytes) -> char* {
    char* p = base + off;
    off += (bytes + 255) & ~(size_t)255;
    return p;
  };
  unsigned int* hist          = (unsigned int*)take(4 * NLVL * 256 * sizeof(unsigned int));
  unsigned int* state         = (unsigned int*)take(32 * sizeof(unsigned int));
  unsigned int* sel           = (unsigned int*)take(MTOT * sizeof(unsigned int));
  unsigned int* eqbuf         = (unsigned int*)take(NLVL * EQCAP * sizeof(unsigned int));
  unsigned long long* keys    = (unsigned long long*)take(MTOT * sizeof(unsigned long long));
  unsigned int* sortedGa      = (unsigned int*)take(MTOT * sizeof(unsigned int));
  float* boxesS               = (float*)take(MTOT * 4 * sizeof(float));
  float4* nbS                 = (float4*)take(MTOT * sizeof(float4));
  float* areasS               = (float*)take(MTOT * sizeof(float));
  unsigned long long* validBits = (unsigned long long*)take(NW * sizeof(unsigned long long));
  unsigned long long* mask    = (unsigned long long*)take((size_t)MTOT * NW * sizeof(unsigned long long));

  k_init<<<64, 256, 0, stream>>>(hist, state, validBits, out);
  for (int p = 0; p < 4; ++p) {
    int shiftB = 24 - 8 * p;
    unsigned int pmask = (p == 0) ? 0u : (0xFFFFFFFFu << (32 - 8 * p));
    unsigned int* histp = hist + p * NLVL * 256;
    k_hist<<<dim3(256, NLVL), 256, 0, stream>>>(obj, state, histp, shiftB, pmask);
    k_select<<<NLVL, 32, 0, stream>>>(histp, state, shiftB);
  }
  k_compact<<<2048, 256, 0, stream>>>(obj, state, sel, eqbuf);
  k_eq<<<NLVL, 256, 0, stream>>>(state, eqbuf, sel);
  k_keys<<<(MTOT + 255) / 256, 256, 0, stream>>>(obj, deltas, anchors, sel, state, keys);
  k_sort<<<1, 1024, 0, stream>>>(keys, sortedGa);
  k_post<<<(MTOT + 255) / 256, 256, 0, stream>>>(deltas, anchors, sortedGa, state,
                                                 boxesS, nbS, areasS, validBits);
  k_mask<<<dim3(NW, NW), 64, 0, stream>>>(nbS, areasS, mask);
  k_reduce<<<1, 32, 0, stream>>>(boxesS, validBits, mask, out);
}